// SingleViewGCN_89378269429931
// MI455X (gfx1250) — compile-verified
//
#include <hip/hip_runtime.h>

typedef float v2f __attribute__((ext_vector_type(2)));
typedef float v8f __attribute__((ext_vector_type(8)));

#define NFEAT 64   // H == OUT == 64
#define BN_EPS 1e-5f

// ---------------- degree / normalization ----------------

__global__ void k_fill(float* __restrict__ p, float v, int n) {
    int i = blockIdx.x * blockDim.x + threadIdx.x;
    if (i < n) p[i] = v;
}

__global__ void k_deg_accum(const int* __restrict__ dst, float* __restrict__ deg, int E) {
    int e = blockIdx.x * blockDim.x + threadIdx.x;
    if (e < E) atomicAdd(&deg[dst[e]], 1.0f);
}

__global__ void k_dinv(float* __restrict__ deg, int n) {
    int i = blockIdx.x * blockDim.x + threadIdx.x;
    if (i < n) deg[i] = rsqrtf(fmaxf(deg[i], 1e-12f));
}

// ---------------- fp32 WMMA GEMM: out[N x 64] = A[N x K] * W[K x 64] ----------------
// One wave computes one 16x16 tile of the output using V_WMMA_F32_16X16X4_F32.
// A-frag: lane l -> row l%16, K pair at 2*(l/16).  B-frag mirrored.
// C/D: vgpr j -> row j + 8*(l/16), col l%16.

__global__ void k_gemm_wmma(const float* __restrict__ A, const float* __restrict__ W,
                            float* __restrict__ out, int nrows, int K) {
    const int NCOL = NFEAT;
    int wave = (int)((blockIdx.x * blockDim.x + threadIdx.x) >> 5);
    int lane = threadIdx.x & 31;
    const int tcols = NCOL / 16;                 // 4 column tiles
    int row0 = (wave / tcols) * 16;              // wave-uniform
    int col0 = (wave % tcols) * 16;
    if (row0 >= nrows) return;                   // whole-wave exit: EXEC stays all-ones

    int half = lane >> 4;                        // 0 or 1
    int l16  = lane & 15;

    const float* arow = A + (size_t)(row0 + l16) * K + 2 * half;
    const float* wcol = W + (size_t)(2 * half) * NCOL + col0 + l16;

    v8f c = {};
    for (int k = 0; k < K; k += 4) {
        v2f a, b;
        a.x = arow[k];
        a.y = arow[k + 1];
        b.x = wcol[(size_t)k * NCOL];
        b.y = wcol[(size_t)(k + 1) * NCOL];
        c = __builtin_amdgcn_wmma_f32_16x16x4_f32(false, a, false, b,
                                                  (short)0, c, false, false);
    }

    float* orow = out + (size_t)(row0 + 8 * half) * NCOL + col0 + l16;
#pragma unroll
    for (int j = 0; j < 8; ++j)
        orow[(size_t)j * NCOL] = c[j];
}

// ---------------- aggregation: out = D^-1/2 (A+I) D^-1/2 h + b ----------------

// self-loop + bias init: out[i][f] = b[f] + h[i][f] * dinv[i]^2
__global__ void k_agg_init(const float* __restrict__ h, const float* __restrict__ dinv,
                           const float* __restrict__ bias, float* __restrict__ out, int n) {
    int idx = blockIdx.x * blockDim.x + threadIdx.x;
    if (idx >= n * NFEAT) return;
    int i = idx >> 6;
    int f = idx & 63;
    float di = dinv[i];
    out[idx] = bias[f] + h[idx] * di * di;
}

// edge scatter-add: out[dst][f] += h[src][f] * dinv[src]*dinv[dst]
__global__ void k_agg_edges(const float* __restrict__ h, const int* __restrict__ src,
                            const int* __restrict__ dst, const float* __restrict__ dinv,
                            float* __restrict__ out, int E) {
    int t = blockIdx.x * blockDim.x + threadIdx.x;
    int e = t >> 6;
    if (e >= E) return;
    int f = t & 63;
    int s = src[e], d = dst[e];
    float w = dinv[s] * dinv[d];
    atomicAdd(&out[(size_t)d * NFEAT + f], h[(size_t)s * NFEAT + f] * w);
}

// ---------------- batch norm ----------------

// partial sums: stats[f] += sum, stats[64+f] += sumsq  (stats pre-zeroed)
__global__ void k_bn_stats(const float* __restrict__ h, float* __restrict__ stats, int n) {
    int f = threadIdx.x & 63;
    int part   = (int)((blockIdx.x * blockDim.x + threadIdx.x) >> 6);
    int nparts = (int)((gridDim.x * blockDim.x) >> 6);
    float s = 0.f, sq = 0.f;
    for (int i = part; i < n; i += nparts) {
        float v = h[(size_t)i * NFEAT + f];
        s += v;
        sq += v * v;
    }
    atomicAdd(&stats[f], s);
    atomicAdd(&stats[NFEAT + f], sq);
}

// scale/shift: sc[f] = gamma*rstd, sc[64+f] = beta - mean*gamma*rstd
__global__ void k_bn_param(const float* __restrict__ stats, const float* __restrict__ gamma,
                           const float* __restrict__ beta, float* __restrict__ sc, int n) {
    int f = threadIdx.x;
    if (f >= NFEAT) return;
    float inv_n = 1.0f / (float)n;
    float mean = stats[f] * inv_n;
    float var  = stats[NFEAT + f] * inv_n - mean * mean;  // biased, matches jnp.var
    float a = gamma[f] * rsqrtf(var + BN_EPS);
    sc[f] = a;
    sc[NFEAT + f] = beta[f] - mean * a;
}

__global__ void k_bn_apply(const float* __restrict__ h, const float* __restrict__ sc,
                           float* __restrict__ out, int n, int do_relu) {
    int idx = blockIdx.x * blockDim.x + threadIdx.x;
    if (idx >= n * NFEAT) return;
    int f = idx & 63;
    float v = fmaf(h[idx], sc[f], sc[NFEAT + f]);
    if (do_relu) v = fmaxf(v, 0.f);
    out[idx] = v;
}

// ---------------- launch ----------------

extern "C" void kernel_launch(void* const* d_in, const int* in_sizes, int n_in,
                              void* d_out, int out_size, void* d_ws, size_t ws_size,
                              hipStream_t stream) {
    const float* x   = (const float*)d_in[0];
    const int*   ei  = (const int*)d_in[1];
    const float* W1  = (const float*)d_in[2];
    const float* b1  = (const float*)d_in[3];
    const float* g1  = (const float*)d_in[4];
    const float* be1 = (const float*)d_in[5];
    const float* W2  = (const float*)d_in[6];
    const float* b2  = (const float*)d_in[7];
    const float* g2  = (const float*)d_in[8];
    const float* be2 = (const float*)d_in[9];

    const int IN = 128;
    const int N  = in_sizes[0] / IN;     // 50000
    const int E  = in_sizes[1] / 2;      // 800000
    const int* src = ei;
    const int* dst = ei + E;

    // workspace carve-up (floats)
    float* ws    = (float*)d_ws;
    float* dinv  = ws;                               // N
    float* bufA  = dinv + N;                         // N*64
    float* bufB  = bufA + (size_t)N * NFEAT;         // N*64
    float* stats = bufB + (size_t)N * NFEAT;         // 128
    float* sc    = stats + 2 * NFEAT;                // 128

    const int TB = 256;
    int nf        = N * NFEAT;
    int grid_n    = (N + TB - 1) / TB;
    int grid_e    = (E + TB - 1) / TB;
    int grid_nf   = (nf + TB - 1) / TB;
    long long et  = (long long)E * NFEAT;
    int grid_ef   = (int)((et + TB - 1) / TB);
    // GEMM: one wave per 16x16 tile; (N/16)*(64/16) waves
    int waves     = ((N + 15) / 16) * (NFEAT / 16);
    int grid_gemm = (waves * 32 + TB - 1) / TB;

    // --- degrees & symmetric norm ---
    k_fill<<<grid_n, TB, 0, stream>>>(dinv, 1.0f, N);          // self loops
    k_deg_accum<<<grid_e, TB, 0, stream>>>(dst, dinv, E);
    k_dinv<<<grid_n, TB, 0, stream>>>(dinv, N);

    // --- layer 1: GCNConv -> BN -> ReLU ---
    k_gemm_wmma<<<grid_gemm, TB, 0, stream>>>(x, W1, bufA, N, IN);
    k_agg_init<<<grid_nf, TB, 0, stream>>>(bufA, dinv, b1, bufB, N);
    k_agg_edges<<<grid_ef, TB, 0, stream>>>(bufA, src, dst, dinv, bufB, E);
    k_fill<<<1, 2 * NFEAT, 0, stream>>>(stats, 0.0f, 2 * NFEAT);
    k_bn_stats<<<256, TB, 0, stream>>>(bufB, stats, N);
    k_bn_param<<<1, NFEAT, 0, stream>>>(stats, g1, be1, sc, N);
    k_bn_apply<<<grid_nf, TB, 0, stream>>>(bufB, sc, bufA, N, 1);

    // --- layer 2: GCNConv -> BN ---
    k_gemm_wmma<<<grid_gemm, TB, 0, stream>>>(bufA, W2, bufB, N, NFEAT);
    k_agg_init<<<grid_nf, TB, 0, stream>>>(bufB, dinv, b2, bufA, N);
    k_agg_edges<<<grid_ef, TB, 0, stream>>>(bufB, src, dst, dinv, bufA, E);
    k_fill<<<1, 2 * NFEAT, 0, stream>>>(stats, 0.0f, 2 * NFEAT);
    k_bn_stats<<<256, TB, 0, stream>>>(bufA, stats, N);
    k_bn_param<<<1, NFEAT, 0, stream>>>(stats, g2, be2, sc, N);
    k_bn_apply<<<grid_nf, TB, 0, stream>>>(bufA, sc, (float*)d_out, N, 0);
}